// VirtualValueNetwork_16063177687585
// MI455X (gfx1250) — compile-verified
//
#include <hip/hip_runtime.h>
#include <hip/hip_bf16.h>

// Problem geometry (fixed by the reference).
#define NUM_BUYERS 1024
#define BATCH_SZ   16384
// 6 coefficients (K=2 x J=3) per buyer.

#define THREADS        256   // 8 wave32s; 256 threads * 4 buyers = 1024 = full row
#define BUY_PER_THR    4
#define ROWS_PER_BLOCK 32    // batch rows streamed per block
#define PREFETCH_ROWS  8

// Native clang vector types: accepted by the nontemporal builtins (HIP float4 is not).
typedef float v4f __attribute__((ext_vector_type(4)));
typedef float v2f __attribute__((ext_vector_type(2)));

// virtual value for one buyer: min over k of (max over j of bid*ew + beta)
__device__ __forceinline__ float vv_one(float bid, const float* __restrict__ ew,
                                        const float* __restrict__ bt) {
    float a0 = fmaf(bid, ew[0], bt[0]);
    float a1 = fmaf(bid, ew[1], bt[1]);
    float a2 = fmaf(bid, ew[2], bt[2]);
    float b0 = fmaf(bid, ew[3], bt[3]);
    float b1 = fmaf(bid, ew[4], bt[4]);
    float b2 = fmaf(bid, ew[5], bt[5]);
    float m0 = fmaxf(fmaxf(a0, a1), a2);   // -> v_max3_f32
    float m1 = fmaxf(fmaxf(b0, b1), b2);
    return fminf(m0, m1);
}

__global__ __launch_bounds__(THREADS)
void virtual_values_kernel(const float* __restrict__ bids,
                           const float* __restrict__ w_params,
                           const float* __restrict__ beta_params,
                           float* __restrict__ out) {
    const int t    = threadIdx.x;
    const int col  = t * BUY_PER_THR;                // first buyer this thread owns
    const int row0 = blockIdx.x * ROWS_PER_BLOCK;    // first batch row this block owns

    // ---- one-time per-block setup: load 4 buyers x 6 coeffs, exp() the weights ----
    float ew[BUY_PER_THR][6];
    float bt[BUY_PER_THR][6];
#pragma unroll
    for (int u = 0; u < BUY_PER_THR; ++u) {
        // per-buyer stride = 6 floats = 24 bytes -> 8-byte aligned: use 64-bit loads
        const v2f* wp = reinterpret_cast<const v2f*>(w_params    + (size_t)(col + u) * 6);
        const v2f* pp = reinterpret_cast<const v2f*>(beta_params + (size_t)(col + u) * 6);
#pragma unroll
        for (int q = 0; q < 3; ++q) {
            v2f wv = wp[q];
            v2f bv = pp[q];
            ew[u][2 * q]     = expf(wv.x);
            ew[u][2 * q + 1] = expf(wv.y);
            bt[u][2 * q]     = bv.x;
            bt[u][2 * q + 1] = bv.y;
        }
    }

    // ---- stream ROWS_PER_BLOCK batch rows: 1 b128 load + 1 b128 store per row ----
    const float* bp = bids + (size_t)row0 * NUM_BUYERS + col;
    float*       op = out  + (size_t)row0 * NUM_BUYERS + col;

    for (int r = 0; r < ROWS_PER_BLOCK; ++r) {
        // keep the read stream ahead of the ALU (emits global_prefetch_b8)
        __builtin_prefetch(bp + (size_t)PREFETCH_ROWS * NUM_BUYERS, 0, 0);

        // single-use streaming data: non-temporal load (TH=NT on global_load_b128)
        v4f b4 = __builtin_nontemporal_load(reinterpret_cast<const v4f*>(bp));

        v4f o4;
        o4.x = vv_one(b4.x, ew[0], bt[0]);
        o4.y = vv_one(b4.y, ew[1], bt[1]);
        o4.z = vv_one(b4.z, ew[2], bt[2]);
        o4.w = vv_one(b4.w, ew[3], bt[3]);

        // streamed output: non-temporal store (TH=NT on global_store_b128)
        __builtin_nontemporal_store(o4, reinterpret_cast<v4f*>(op));

        bp += NUM_BUYERS;
        op += NUM_BUYERS;
    }
}

extern "C" void kernel_launch(void* const* d_in, const int* in_sizes, int n_in,
                              void* d_out, int out_size, void* d_ws, size_t ws_size,
                              hipStream_t stream) {
    const float* bids        = (const float*)d_in[0];   // [16384, 1024]
    const float* w_params    = (const float*)d_in[1];   // [1024, 2, 3]
    const float* beta_params = (const float*)d_in[2];   // [1024, 2, 3]
    float*       out         = (float*)d_out;           // [16384, 1024]

    (void)in_sizes; (void)n_in; (void)out_size; (void)d_ws; (void)ws_size;

    dim3 grid(BATCH_SZ / ROWS_PER_BLOCK);  // 512 blocks
    dim3 block(THREADS);                   // 256 threads = 8 wave32s
    virtual_values_kernel<<<grid, block, 0, stream>>>(bids, w_params, beta_params, out);
}